// IntraAgg_20083267076375
// MI455X (gfx1250) — compile-verified
//
#include <hip/hip_runtime.h>
#include <stdint.h>

typedef __attribute__((ext_vector_type(2))) float v2f;
typedef __attribute__((ext_vector_type(8))) float v8f;

#define D_FEAT          128
#define K_NEIGH         32
#define WAVES_PER_BLOCK 2
#define ROW_F           132              // padded LDS row: 132 floats = 528 B (16B aligned, bank-skewed)
#define ROWS_PER_WAVE   33               // 1 center + 32 neighbors
#define THREADS         (WAVES_PER_BLOCK * 32)

__global__ __launch_bounds__(THREADS) void intra_agg_kernel(
    const float* __restrict__ features,   // [N_NODES, 128]
    const int*   __restrict__ nodes,      // [B]
    const int*   __restrict__ neighs,     // [B, 32]
    const int*   __restrict__ num_sample_p,
    float*       __restrict__ out,        // [B, 128]
    int batch)
{
    __shared__ float lds[WAVES_PER_BLOCK * ROWS_PER_WAVE * ROW_F];

    const int lane = threadIdx.x & 31;
    const int wave = threadIdx.x >> 5;
    const int b    = blockIdx.x * WAVES_PER_BLOCK + wave;
    if (b >= batch) return;                       // wave-uniform; EXEC stays all-ones

    float* wbase = &lds[wave * ROWS_PER_WAVE * ROW_F];
    // Low 32 bits of a generic LDS pointer == wave-relative LDS byte offset.
    const uint32_t lds0 = (uint32_t)(uintptr_t)wbase;
    const uint64_t fbase = (uint64_t)(uintptr_t)features;

    // ---------------- Stage 33 rows into LDS with async copies (ASYNCcnt) ----------------
    {
        const int node = nodes[b];
        uint64_t g = fbase + ((uint64_t)(uint32_t)node << 9) + (uint32_t)(lane << 4);
        uint32_t l = lds0 + (uint32_t)(lane << 4);
        asm volatile("global_load_async_to_lds_b128 %0, %1, off" :: "v"(l), "v"(g) : "memory");
    }
    const int my_nid = neighs[b * K_NEIGH + lane];
#pragma unroll
    for (int k = 0; k < K_NEIGH; ++k) {
        int nid = __shfl(my_nid, k, 32);
        uint64_t g = fbase + ((uint64_t)(uint32_t)nid << 9) + (uint32_t)(lane << 4);
        uint32_t l = lds0 + (uint32_t)((1 + k) * ROW_F * 4) + (uint32_t)(lane << 4);
        asm volatile("global_load_async_to_lds_b128 %0, %1, off" :: "v"(l), "v"(g) : "memory");
    }
    asm volatile("s_wait_asynccnt 0x0" ::: "memory");

    // ---------------- Dots via V_WMMA_F32_16X16X4_F32 ----------------
    // A[m][k] = center[d+k] for all m (broadcast rows); B[k][n] = neigh[n_global][d+k].
    // => D[m][n] = sum_k center[d+k]*neigh[n][d+k], replicated over m.
    // f32 A/B layout: lanes 0-15 hold K=(0,1), lanes 16-31 hold K=(2,3)  -> one 8B LDS load each.
    const int hi  = lane >> 4;        // 0 | 1
    const int nlo = lane & 15;
    const float* crow = wbase;                                 // center row
    const float* r0   = wbase + (1 + nlo) * ROW_F;             // neighbor group 0 column
    const float* r1   = wbase + (1 + 16 + nlo) * ROW_F;        // neighbor group 1 column
    v8f accA = {}; v8f accB = {};
#pragma unroll
    for (int d = 0; d < D_FEAT; d += 4) {
        v2f a  = *(const v2f*)(crow + d + 2 * hi);
        v2f b0 = *(const v2f*)(r0   + d + 2 * hi);
        v2f b1 = *(const v2f*)(r1   + d + 2 * hi);
        accA = __builtin_amdgcn_wmma_f32_16x16x4_f32(false, a, false, b0, (short)0, accA, false, false);
        accB = __builtin_amdgcn_wmma_f32_16x16x4_f32(false, a, false, b1, (short)0, accB, false, false);
    }
    // D VGPR0: lanes 0-15 = D[0][lane] (neighbor=lane); lanes 16-31 = D[8][lane-16] (neighbor=lane).
    float dot = (lane < 16) ? accA[0] : accB[0];

    // ---------------- Self norm of my neighbor row (VALU, co-executes with WMMA tail) --------
    const float* myrow = wbase + (1 + lane) * ROW_F;
    float nn = 0.f;
#pragma unroll
    for (int d = 0; d < D_FEAT; d += 4) {
        float4 v = *(const float4*)(myrow + d);
        nn += v.x * v.x + v.y * v.y + v.z * v.z + v.w * v.w;
    }
    // Center norm is a common positive factor across the 32 candidates: cannot change top-k order.
    float r = dot * rsqrtf(nn);

    // ---------------- Rank (ties -> lower index first, matches lax.top_k) ----------------
    int rank = 0;
#pragma unroll
    for (int j = 0; j < 32; ++j) {
        float rj = __shfl(r, j, 32);
        rank += ((rj > r) || (rj == r && j < lane)) ? 1 : 0;
    }
    const int ns = num_sample_p[0];
    unsigned mask = (unsigned)__ballot(rank < ns);
    int cnt = __popc(mask);

    // ---------------- Masked mean + ReLU; lane owns dims [4*lane, 4*lane+4) ----------------
    float4 acc = make_float4(0.f, 0.f, 0.f, 0.f);
    unsigned m = mask;                       // wave-uniform: no divergence
    while (m) {
        int k = __ffs(m) - 1;
        m &= m - 1;
        const float4 v = *(const float4*)(wbase + (1 + k) * ROW_F + 4 * lane);
        acc.x += v.x; acc.y += v.y; acc.z += v.z; acc.w += v.w;
    }
    float inv = (cnt > 0) ? (1.0f / (float)cnt) : 0.0f;
    float4 res;
    res.x = fmaxf(acc.x * inv, 0.f);
    res.y = fmaxf(acc.y * inv, 0.f);
    res.z = fmaxf(acc.z * inv, 0.f);
    res.w = fmaxf(acc.w * inv, 0.f);
    *((float4*)(out + (size_t)b * D_FEAT) + lane) = res;       // coalesced b128 store
}

extern "C" void kernel_launch(void* const* d_in, const int* in_sizes, int n_in,
                              void* d_out, int out_size, void* d_ws, size_t ws_size,
                              hipStream_t stream) {
    const float* features   = (const float*)d_in[0];
    const int*   nodes      = (const int*)d_in[1];
    const int*   neighs     = (const int*)d_in[2];
    const int*   num_sample = (const int*)d_in[3];
    float*       out        = (float*)d_out;

    const int batch = in_sizes[1];                 // nodes: [B]
    const int grid  = (batch + WAVES_PER_BLOCK - 1) / WAVES_PER_BLOCK;
    hipLaunchKernelGGL(intra_agg_kernel, dim3(grid), dim3(THREADS), 0, stream,
                       features, nodes, neighs, num_sample, out, batch);
}